// DigitLayer_33526514712963
// MI455X (gfx1250) — compile-verified
//
#include <hip/hip_runtime.h>

typedef float v2f __attribute__((ext_vector_type(2)));
typedef float v8f __attribute__((ext_vector_type(8)));

#define B_   256
#define P_   1152
#define D_   10
#define VP_  8
#define VD_  16
#define KTOT (P_ * VP_)          // 9216
#define NWAVE 8
#define P_PER_WAVE (P_ / NWAVE)  // 144

// One workgroup per (m-tile of 16 batches) x (one capsule d).
// GEMM: S[16 x 16] = X_tile[16 x 9216] @ Wcol[9216 x 16] via v_wmma_f32_16x16x4_f32,
// K split across 8 waves, LDS cross-wave reduce, fused squash.
__global__ __launch_bounds__(256) void caps_wmma_kernel(
    const float* __restrict__ x,    // [B, P, 8, 1]
    const float* __restrict__ W,    // [P, D, 16, 8]
    float* __restrict__ out)        // [B, D, 16, 1]
{
  __shared__ float lds_part[NWAVE * 32 * 8];   // per-wave v8f partials
  __shared__ float sred[16][17];               // reduced S tile (padded)
  __shared__ float sscale[16];                 // squash scale per row

  const int tile = blockIdx.x;        // 0..159
  const int mt   = tile / D_;         // batch tile: rows mt*16 .. mt*16+15
  const int d    = tile % D_;         // capsule index
  const int tid  = threadIdx.x;
  const int lane = tid & 31;
  const int wave = tid >> 5;
  const int half = lane >> 4;         // selects K pair {0,1} vs {2,3}
  const int sub  = lane & 15;         // M row (A) / N col i (B)

  // A operand: x[b = mt*16+sub, k], k-pair at +half*2 within each 4-step
  const float* xrow = x + (size_t)(mt * 16 + sub) * KTOT + half * 2;
  // B operand: Wmat[k, n=d*16+sub] = W[p*1280 + d*128 + sub*8 + j], j = k%8
  const float* wcol = W + (size_t)d * (VD_ * VP_) + (size_t)sub * VP_ + half * 2;

  v8f acc0 = {};   // k = 8p + 0..3 chain
  v8f acc1 = {};   // k = 8p + 4..7 chain (independent for ILP)

  const int pbeg = wave * P_PER_WAVE;
  #pragma unroll 4
  for (int p = pbeg; p < pbeg + P_PER_WAVE; ++p) {
    const float* xp = xrow + p * VP_;
    const float* wp = wcol + (size_t)p * (D_ * VD_ * VP_);
    v2f a0 = *(const v2f*)(xp);
    v2f b0 = *(const v2f*)(wp);
    v2f a1 = *(const v2f*)(xp + 4);
    v2f b1 = *(const v2f*)(wp + 4);
    acc0 = __builtin_amdgcn_wmma_f32_16x16x4_f32(
        false, a0, false, b0, (short)0, acc0, false, false);
    acc1 = __builtin_amdgcn_wmma_f32_16x16x4_f32(
        false, a1, false, b1, (short)0, acc1, false, false);
  }

  v8f acc = acc0 + acc1;

  // stash per-wave partial C tiles
  #pragma unroll
  for (int r = 0; r < 8; ++r)
    lds_part[(wave * 32 + lane) * 8 + r] = acc[r];
  __syncthreads();

  // cross-wave reduction: 256 threads cover all (lane, vgpr) combos
  {
    const int li = tid & 31;
    const int r  = tid >> 5;
    float s = 0.f;
    #pragma unroll
    for (int w = 0; w < NWAVE; ++w)
      s += lds_part[(w * 32 + li) * 8 + r];
    // C layout: M = r + (lane>=16 ? 8 : 0), N = lane & 15
    const int M = r + ((li >> 4) << 3);
    const int N = li & 15;
    sred[M][N] = s;
  }
  __syncthreads();

  // squash scale per batch row: sq/(1+sq)/sqrt(sq+1e-9)
  if (tid < 16) {
    float sq = 0.f;
    #pragma unroll
    for (int i = 0; i < 16; ++i) { float v = sred[tid][i]; sq += v * v; }
    sscale[tid] = sq / (1.0f + sq) / sqrtf(sq + 1e-9f);
  }
  __syncthreads();

  // write v[b, d, i] = S * scale ; out layout [B, D, 16, 1]
  {
    const int M = tid >> 4;
    const int N = tid & 15;
    out[(size_t)(mt * 16 + M) * (D_ * VD_) + d * VD_ + N] = sred[M][N] * sscale[M];
  }
}

extern "C" void kernel_launch(void* const* d_in, const int* in_sizes, int n_in,
                              void* d_out, int out_size, void* d_ws, size_t ws_size,
                              hipStream_t stream) {
  const float* x   = (const float*)d_in[0];   // [256,1152,8,1]
  const float* W   = (const float*)d_in[1];   // [1152,10,16,8]
  float*       out = (float*)d_out;           // [256,10,16,1]
  (void)in_sizes; (void)n_in; (void)out_size; (void)d_ws; (void)ws_size;

  dim3 grid((B_ / 16) * D_);   // 160 tiles
  caps_wmma_kernel<<<grid, 256, 0, stream>>>(x, W, out);
}